// FieldWeightedFactorizationMachine_36653250904721
// MI455X (gfx1250) — compile-verified
//
#include <hip/hip_runtime.h>

#define NF      39     // NUM_FACTORS
#define EMBD    64     // EMB_DIM
#define TILES   3      // ceil(39/16)
#define PADF    48     // padded factor count
#define ESTRIDE 68     // LDS row stride in floats (bank-conflict-free frag loads)
#define WAVES   4      // waves (=samples) per block

typedef __attribute__((ext_vector_type(2))) float v2f;
typedef __attribute__((ext_vector_type(8))) float v8f;

static __device__ __forceinline__ v8f wmma_f32_k4(v2f a, v2f b, v8f c) {
  // D(16x16,f32) = A(16x4,f32) x B(4x16,f32) + C
  return __builtin_amdgcn_wmma_f32_16x16x4_f32(
      /*neg_a=*/false, a, /*neg_b=*/false, b,
      /*c_mod=*/(short)0, c, /*reuse_a=*/false, /*reuse_b=*/false);
}

__global__ __launch_bounds__(WAVES * 32)
void fwfm_wmma_kernel(const int* __restrict__ x,        // (B, 39) indices
                      const float* __restrict__ w0,     // (1,)
                      const float* __restrict__ bias,   // (1M, 1)
                      const float* __restrict__ emb,    // (1M, 64)
                      const float* __restrict__ W,      // (39, 39)
                      float* __restrict__ out,          // (B,)
                      int B) {
  __shared__ float E[WAVES][PADF * ESTRIDE];  // gathered embeddings, one tile per wave
  __shared__ float Wm[PADF * PADF];           // strictly-upper-masked W, zero padded

  const int tid  = threadIdx.x;
  const int wv   = tid >> 5;
  const int lane = tid & 31;
  const int b    = blockIdx.x * WAVES + wv;
  const bool active = (b < B);

  // ---- Stage masked W once per block: Wm[i][j] = (i<j<39) ? W[i,j] : 0 ----
  for (int t = tid; t < PADF * PADF; t += WAVES * 32) {
    int i = t / PADF, j = t - i * PADF;
    float w = 0.0f;
    if (i < j && j < NF) w = W[i * NF + j];
    Wm[t] = w;
  }

  float partial = 0.0f;
  if (active) {
    const int* xr = x + b * NF;
    // ---- Gather 39 embedding rows into LDS (float4 vectorized) ----
    for (int t = lane; t < NF * (EMBD / 4); t += 32) {
      int row = t >> 4, c4 = t & 15;
      int idx = xr[row];                    // tiny hot region, cache hit
      const float4 v = *(const float4*)(emb + (size_t)idx * EMBD + c4 * 4);
      *(float4*)(&E[wv][row * ESTRIDE + c4 * 4]) = v;
    }
    // ---- Zero-fill padded rows 39..47 (avoid NaN garbage feeding WMMA) ----
    float4 z; z.x = z.y = z.z = z.w = 0.0f;
    for (int t = lane; t < (PADF - NF) * (EMBD / 4); t += 32) {
      int row = NF + (t >> 4), c4 = t & 15;
      *(float4*)(&E[wv][row * ESTRIDE + c4 * 4]) = z;
    }
    // ---- Bias partial sum ----
    for (int f = lane; f < NF; f += 32) partial += bias[xr[f]];
  }
  __syncthreads();

  if (active) {
    const float* Ew = E[wv];
    const int m    = lane & 15;            // row within tile (A) / col within tile (B)
    const int koff = (lane >> 4) << 1;     // K sub-offset for upper half-wave

    // Gram tiles G[I][J] for J>=I (symmetric; strict-lower masked to zero anyway)
    v8f c00 = {}, c01 = {}, c02 = {}, c11 = {}, c12 = {}, c22 = {};
    for (int s = 0; s < EMBD / 4; ++s) {
      const int kb = 4 * s + koff;
      // A-frag(16x4) and B-frag(4x16) of the Gram coincide per ISA layout:
      // lane holds E[tile*16 + (lane&15)][kb], E[...][kb+1]
      v2f a0 = *(const v2f*)(&Ew[(0 * 16 + m) * ESTRIDE + kb]);
      v2f a1 = *(const v2f*)(&Ew[(1 * 16 + m) * ESTRIDE + kb]);
      v2f a2 = *(const v2f*)(&Ew[(2 * 16 + m) * ESTRIDE + kb]);
      c00 = wmma_f32_k4(a0, a0, c00);
      c01 = wmma_f32_k4(a0, a1, c01);
      c02 = wmma_f32_k4(a0, a2, c02);
      c11 = wmma_f32_k4(a1, a1, c11);
      c12 = wmma_f32_k4(a1, a2, c12);
      c22 = wmma_f32_k4(a2, a2, c22);
    }

    // ---- Epilogue: acc += Wm[gi][gj] * G[gi][gj] using C/D VGPR layout ----
    // C/D 16x16 f32: lanes 0-15 -> M=r, N=lane; lanes 16-31 -> M=r+8, N=lane-16
    const int i0 = (lane >> 4) << 3;
    const int n  = lane & 15;
    float acc = partial;
#pragma unroll
    for (int r = 0; r < 8; ++r) {
      const int gi = i0 + r;
      acc += Wm[gi * PADF + n]             * c00[r];
      acc += Wm[gi * PADF + 16 + n]        * c01[r];
      acc += Wm[gi * PADF + 32 + n]        * c02[r];
      acc += Wm[(16 + gi) * PADF + 16 + n] * c11[r];
      acc += Wm[(16 + gi) * PADF + 32 + n] * c12[r];
      acc += Wm[(32 + gi) * PADF + 32 + n] * c22[r];
    }

    // ---- wave32 reduction ----
    for (int off = 16; off > 0; off >>= 1)
      acc += __shfl_xor(acc, off, 32);
    if (lane == 0) out[b] = w0[0] + acc;
  }
}

extern "C" void kernel_launch(void* const* d_in, const int* in_sizes, int n_in,
                              void* d_out, int out_size, void* d_ws, size_t ws_size,
                              hipStream_t stream) {
  (void)n_in; (void)d_ws; (void)ws_size; (void)out_size;
  // setup_inputs order: x, w0, bias_table, emb_table, W
  const int*   x    = (const int*)d_in[0];   // integer input -> const int* (harness convention)
  const float* w0   = (const float*)d_in[1];
  const float* bias = (const float*)d_in[2];
  const float* emb  = (const float*)d_in[3];
  const float* W    = (const float*)d_in[4];
  float* out = (float*)d_out;

  const int B = in_sizes[0] / NF;  // 16384
  dim3 block(WAVES * 32);
  dim3 grid((B + WAVES - 1) / WAVES);
  fwfm_wmma_kernel<<<grid, block, 0, stream>>>(x, w0, bias, emb, W, out, B);
}